// WavePool3D_21053929685533
// MI455X (gfx1250) — compile-verified
//
#include <hip/hip_runtime.h>

// CDNA5 / gfx1250: wave32, WMMA 16x16x4 f32, ds_bpermute lane merge.
typedef __attribute__((ext_vector_type(2))) float v2f;
typedef __attribute__((ext_vector_type(8))) float v8f;

// x: [1, 32, 128, 128, 128] f32
// out: 8 Haar subbands concat, each [1, 32, 64, 64, 64] f32
//
// Per 16-block tile:  D = A1*B1 + A2*B2  (V_WMMA_F32_16X16X4_F32, K split over depth planes)
//   A (16x4): Haar coeffs, rows = filter f (rows 8..15 zero), K' = j*2+m
//   B (4x16): rows 0,1 -> VGPR0/1 lanes 0-15 (row 2h), rows 2,3 -> lanes 16-31 (row 2h+1)
//             == exactly one global_load_b64 per depth plane, zero shuffles.
// Tiles are processed in pairs; D2's lanes 0-15 are bpermuted into lanes 16-31 so each
// subband store is a full-wave 128B contiguous non-temporal store.

__device__ __forceinline__ float bperm16(float v, int idx4) {
  return __int_as_float(__builtin_amdgcn_ds_bpermute(idx4, __float_as_int(v)));
}

__global__ __launch_bounds__(256) void haar3d_wmma_kernel(const float* __restrict__ x,
                                                          float* __restrict__ out) {
  const int lane = threadIdx.x & 31;
  const int wid  = blockIdx.x * 8 + (threadIdx.x >> 5);   // 0 .. 131071 = 32*64*64 - 1

  const int h = wid & 63;
  const int d = (wid >> 6) & 63;
  const int c = wid >> 12;

  const int half = lane >> 4;   // 0: B rows k'=0,1 (j=0) | 1: rows k'=2,3 (j=1)
  const int f    = lane & 15;   // A-matrix row (filter index)

  // Haar coeff: F[f,i,j,m] = (1/(2*sqrt2)) * s(f2,i)*s(f1,j)*s(f0,m),
  // s(bit,t) = (bit && t==0) ? -1 : +1
  const float c8 = 0.35355339059327373f;
  const float z  = (f < 8) ? c8 : 0.0f;
  const float sI = (f & 4) ? -1.0f : 1.0f;                 // s(f2, i=0); i=1 -> +1
  const float sJ = ((f & 2) && half == 0) ? -1.0f : 1.0f;  // s(f1, j=half)
  const float sM = (f & 1) ? -1.0f : 1.0f;                 // s(f0, m=0); m=1 -> +1

  v2f a1, a2;                       // A VGPR0 = K'(m=0), VGPR1 = K'(m=1)
  a1.x = z * sI * sJ * sM;  a1.y = z * sI * sJ;   // depth plane i=0
  a2.x = z * sJ * sM;       a2.y = z * sJ;        // depth plane i=1

  // Input row base: x[c][2d][2h + half][*] ; +PLANE = depth plane i=1
  const size_t PLANE = 128u * 128u;
  const float* row0 = x + (size_t)c * (PLANE * 128u)
                        + (size_t)(2 * d) * PLANE
                        + (size_t)(2 * h + half) * 128u;

  // Output base pointers, one per subband (hoisted: computed once)
  const size_t SUB   = 32ull * 64u * 64u * 64u;
  const size_t obase = (size_t)c * (64ull * 64u * 64u)
                     + (size_t)d * (64u * 64u)
                     + (size_t)h * 64u + (size_t)lane;
  float* optr[8];
  #pragma unroll
  for (int r = 0; r < 8; ++r) optr[r] = out + obase + (size_t)r * SUB;

  const bool lo    = (lane < 16);
  const int  idx4  = (lane & 15) * 4;        // ds_bpermute byte index: fetch lane&15
  const int  col   = lane & 15;

  #pragma unroll
  for (int p = 0; p < 2; ++p) {              // pair p -> output cols 32p .. 32p+31
    const float* q1 = row0 + 2 * (32 * p + col);        // tile 2p
    const float* q2 = q1 + 32;                          // tile 2p+1 (cols +16)

    v2f b1a = __builtin_nontemporal_load((const v2f*)(q1));
    v2f b1b = __builtin_nontemporal_load((const v2f*)(q1 + PLANE));
    v2f b2a = __builtin_nontemporal_load((const v2f*)(q2));
    v2f b2b = __builtin_nontemporal_load((const v2f*)(q2 + PLANE));

    v8f d1 = {0.f, 0.f, 0.f, 0.f, 0.f, 0.f, 0.f, 0.f};
    d1 = __builtin_amdgcn_wmma_f32_16x16x4_f32(false, a1, false, b1a, (short)0, d1, false, false);
    d1 = __builtin_amdgcn_wmma_f32_16x16x4_f32(false, a2, false, b1b, (short)0, d1, false, false);

    v8f d2 = {0.f, 0.f, 0.f, 0.f, 0.f, 0.f, 0.f, 0.f};
    d2 = __builtin_amdgcn_wmma_f32_16x16x4_f32(false, a1, false, b2a, (short)0, d2, false, false);
    d2 = __builtin_amdgcn_wmma_f32_16x16x4_f32(false, a2, false, b2b, (short)0, d2, false, false);

    // Merge: lane L<16 -> d1[r][L], L>=16 -> d2[r][L-16]; then full-wave 128B NT store.
    #pragma unroll
    for (int r = 0; r < 8; ++r) {
      float v2r = bperm16(d2[r], idx4);      // all lanes read d2[r] from lane (L&15)
      float v   = lo ? d1[r] : v2r;
      __builtin_nontemporal_store(v, optr[r] + 32 * p);
    }
  }
}

extern "C" void kernel_launch(void* const* d_in, const int* in_sizes, int n_in,
                              void* d_out, int out_size, void* d_ws, size_t ws_size,
                              hipStream_t stream) {
  (void)in_sizes; (void)n_in; (void)d_ws; (void)ws_size; (void)out_size;
  const float* x = (const float*)d_in[0];
  float* out = (float*)d_out;
  // 32*64*64 = 131072 waves, 8 waves (256 threads) per block -> 16384 blocks
  haar3d_wmma_kernel<<<dim3(16384), dim3(256), 0, stream>>>(x, out);
}